// CombinedCriterionAE_11879879541054
// MI455X (gfx1250) — compile-verified
//
#include <hip/hip_runtime.h>
#include <math.h>

typedef __attribute__((ext_vector_type(2))) float v2f;
typedef __attribute__((ext_vector_type(8))) float v8f;

// ---------------------------------------------------------------------------
// ws layout: ws[0] = sum of squared point errors, ws[1] = sum of cos(theta)
// ---------------------------------------------------------------------------
__global__ void cc_init_ws(float* ws) {
    if (threadIdx.x < 2) ws[threadIdx.x] = 0.0f;
}

// One wave (32 lanes) owns 16 pred rows; scans all gt points in 16-col tiles.
// dot products via V_WMMA_F32_16X16X4_F32 (K=3 padded to 4 with zeros).
__global__ __launch_bounds__(256) void cc_chamfer_wmma(
    const float* __restrict__ pred,   // Np x 6  (xyz, nxyz)
    const float* __restrict__ gt,     // Ng x 6  (xyz, nxyz)
    float* __restrict__ ws,
    int Ng) {
    const int lane    = threadIdx.x & 31;
    const int wave    = blockIdx.x * (blockDim.x >> 5) + (threadIdx.x >> 5);
    const int rowBase = wave << 4;            // first of this wave's 16 pred rows
    const int col     = lane & 15;
    const bool hi     = lane >= 16;

    // A fragment: 16x4 f32 (M x K). ISA layout: lanes 0-15 -> K=0,1 in v0,v1;
    // lanes 16-31 -> K=2,3. We pad K=3 with 0.
    const float* prow = pred + 6 * (rowBase + col);
    v2f a;
    a.x = hi ? prow[2] : prow[0];
    a.y = hi ? 0.0f    : prow[1];

    float bestV[8];
    int   bestI[8];
#pragma unroll
    for (int v = 0; v < 8; ++v) { bestV[v] = 3.4e38f; bestI[v] = 0; }

    const int ntiles = Ng >> 4;
#pragma unroll 2
    for (int t = 0; t < ntiles; ++t) {
        const int n = (t << 4) + col;        // this lane's gt column
        const float* g = gt + 6 * n;         // 24-byte row stride
        const float gx = g[0], gy = g[1], gz = g[2];
        const float g2 = gx * gx + gy * gy + gz * gz;

        // B fragment: 4x16 f32 (K x N), mirrored layout of A.
        v2f b;
        b.x = hi ? gz   : gx;
        b.y = hi ? 0.0f : gy;

        v8f c = {};
        c = __builtin_amdgcn_wmma_f32_16x16x4_f32(
                /*neg_a=*/false, a, /*neg_b=*/false, b,
                /*c_mod=*/(short)0, c, /*reuse_a=*/false, /*reuse_b=*/false);

        // D layout: lane holds column (lane&15); VGPR v -> row v + 8*hi.
        // argmin_n (g2 - 2*dot)  ==  argmin_n dist^2   (p2 row-constant)
#pragma unroll
        for (int v = 0; v < 8; ++v) {
            const float score = fmaf(-2.0f, c[v], g2);
            const bool  lt    = score < bestV[v];   // strict: keep first index
            bestV[v] = lt ? score : bestV[v];
            bestI[v] = lt ? n     : bestI[v];
        }
    }

    // Cross-lane min+argmin within each 16-lane half (xor masks stay in-half).
#pragma unroll
    for (int v = 0; v < 8; ++v) {
        float val = bestV[v];
        int   idx = bestI[v];
#pragma unroll
        for (int off = 1; off < 16; off <<= 1) {
            const float ov = __shfl_xor(val, off, 32);
            const int   oi = __shfl_xor(idx, off, 32);
            const bool take = (ov < val) || (ov == val && oi < idx);
            val = take ? ov : val;
            idx = take ? oi : idx;
        }
        bestV[v] = val;
        bestI[v] = idx;
    }

    // lane 0 holds argmin for rows rowBase+0..7; lane 16 for rowBase+8..15.
    if (lane == 0 || lane == 16) {
        const int mofs = hi ? 8 : 0;
        float ssq = 0.0f, scos = 0.0f;
        for (int v = 0; v < 8; ++v) {
            const int m = rowBase + mofs + v;
            const int j = bestI[v];
            const float* p = pred + 6 * m;
            const float* g = gt   + 6 * j;
            const float dx = p[0] - g[0], dy = p[1] - g[1], dz = p[2] - g[2];
            ssq += dx * dx + dy * dy + dz * dz;
            const float pnx = p[3], pny = p[4], pnz = p[5];
            const float gnx = g[3], gny = g[4], gnz = g[5];
            const float pn = sqrtf(pnx * pnx + pny * pny + pnz * pnz);
            const float gn = sqrtf(gnx * gnx + gny * gny + gnz * gnz);
            const float invp = 1.0f / fmaxf(pn, 1.0e-4f);
            const float invg = 1.0f / fmaxf(gn, 1.0e-4f);
            scos += (pnx * gnx + pny * gny + pnz * gnz) * invp * invg;
        }
        atomicAdd(&ws[0], ssq);
        atomicAdd(&ws[1], scos);
    }
}

__global__ void cc_finalize(const float* __restrict__ ws,
                            float* __restrict__ out, int Np) {
    if (threadIdx.x == 0 && blockIdx.x == 0) {
        const float inlier = ws[0] / (float)(Np * 3);
        const float nrm    = 1.0f - ws[1] / (float)Np;
        out[0] = inlier + nrm;
    }
}

extern "C" void kernel_launch(void* const* d_in, const int* in_sizes, int n_in,
                              void* d_out, int out_size, void* d_ws, size_t ws_size,
                              hipStream_t stream) {
    // setup_inputs order: pred_feat(8192x6), pred_decoder(unused),
    //                     input_data(unused), gt_data(32768x6)
    const float* pred = (const float*)d_in[0];
    const float* gt   = (const float*)d_in[3];
    float* out = (float*)d_out;
    float* ws  = (float*)d_ws;

    const int Np = in_sizes[0] / 6;   // 8192
    const int Ng = in_sizes[3] / 6;   // 32768

    cc_init_ws<<<1, 32, 0, stream>>>(ws);

    const int waves   = Np / 16;                   // 512 waves, one per 16 rows
    const int threads = 256;                       // 8 waves per block
    const int blocks  = (waves * 32 + threads - 1) / threads;
    cc_chamfer_wmma<<<blocks, threads, 0, stream>>>(pred, gt, ws, Ng);

    cc_finalize<<<1, 32, 0, stream>>>(ws, out, Np);
}